// RegionProposalNetwork_6519760355367
// MI455X (gfx1250) — compile-verified
//
#include <hip/hip_runtime.h>
#include <math.h>

// ---------------- problem constants (must match reference) ----------------
#define NIMG   8
#define ATOT   242991
#define KPRE   4741          // sum over levels of min(n, 1000)
#define POSTN  1000
#define SORTN  8192          // next pow2 >= KPRE
#define NMS_TH 0.7f
#define BBOX_CLIPV 4.135166556742356f   // log(1000/16)
#define IMGW   1216.0f
#define IMGH   800.0f

// bitmask NMS geometry
#define CWORDS 75            // ceil(KPRE/64) 64-bit words per mask row
#define WPC    15            // mask words per column-chunk
#define NCHUNK 5             // WPC*NCHUNK == CWORDS
#define RTILE  256           // rows per mask block

__constant__ int c_lvl_n[5]    = {182400, 45600, 11400, 2850, 741};
__constant__ int c_lvl_off[5]  = {0, 182400, 228000, 239400, 242250};
__constant__ int c_lvl_k[5]    = {1000, 1000, 1000, 1000, 741};
__constant__ int c_slot_off[5] = {0, 1000, 2000, 3000, 4000};

// monotone float -> uint mapping (larger key == larger float, handles -inf)
__device__ __forceinline__ unsigned f2key(float f) {
  unsigned u = __float_as_uint(f);
  return (u & 0x80000000u) ? ~u : (u | 0x80000000u);
}

// ============================================================================
// Kernel 1: per-(image,level) top-k via 4-pass radix select + compaction.
// ============================================================================
__global__ __launch_bounds__(1024) void rpn_topk(const float* __restrict__ obj,
                                                 float* __restrict__ selScore,
                                                 int*   __restrict__ selIdx) {
  const int img = blockIdx.x / 5, lvl = blockIdx.x % 5;
  const int n = c_lvl_n[lvl], off = c_lvl_off[lvl], k = c_lvl_k[lvl];
  const float* sc = obj + (size_t)img * ATOT + off;
  const int tid = threadIdx.x, nt = blockDim.x;

  __shared__ unsigned hist[256];
  __shared__ unsigned sh_prefix, sh_rem, sh_G, sh_cntG, sh_cntT;

  if (tid == 0) { sh_prefix = 0u; sh_rem = (unsigned)k; sh_G = 0u; }
  __syncthreads();

  for (int pass = 0; pass < 4; ++pass) {
    const int shift = 24 - pass * 8;
    const unsigned highMask = (pass == 0) ? 0u : (0xFFFFFFFFu << (shift + 8));
    for (int b = tid; b < 256; b += nt) hist[b] = 0u;
    __syncthreads();
    const unsigned prefix = sh_prefix;
    for (int i = tid; i < n; i += nt) {
      __builtin_prefetch(sc + i + 8 * nt, 0, 1);   // global_prefetch_b8
      unsigned key = f2key(sc[i]);
      if ((key & highMask) == prefix) atomicAdd(&hist[(key >> shift) & 0xFFu], 1u);
    }
    __syncthreads();
    if (tid == 0) {
      unsigned rem = sh_rem, cum = 0u;
      int b;
      for (b = 255; b > 0; --b) {
        unsigned c = hist[b];
        if (cum + c >= rem) break;
        cum += c;
      }
      sh_prefix = prefix | ((unsigned)b << shift);
      sh_G += cum;
      sh_rem = rem - cum;
    }
    __syncthreads();
  }

  const unsigned T = sh_prefix, G = sh_G, tie = sh_rem;
  if (tid == 0) { sh_cntG = 0u; sh_cntT = 0u; }
  __syncthreads();

  float* oS = selScore + (size_t)img * KPRE + c_slot_off[lvl];
  int*   oI = selIdx   + (size_t)img * KPRE + c_slot_off[lvl];
  for (int i = tid; i < n; i += nt) {
    float f = sc[i];
    unsigned key = f2key(f);
    if (key > T) {
      unsigned p = atomicAdd(&sh_cntG, 1u);
      oS[p] = f; oI[p] = off + i;
    } else if (key == T) {
      unsigned t = atomicAdd(&sh_cntT, 1u);
      if (t < tie) { unsigned p = G + t; oS[p] = f; oI[p] = off + i; }
    }
  }
}

// ============================================================================
// Kernel 2: per-image descending bitonic sort in LDS (keys staged by CDNA5
// async global->LDS copies), then decode + clip + batched-NMS level offset.
// ============================================================================
__global__ __launch_bounds__(1024) void rpn_sort_decode(
    const float* __restrict__ selScore, const int* __restrict__ selIdx,
    const float* __restrict__ deltas,   const float* __restrict__ anchors,
    float4* __restrict__ bnms, float4* __restrict__ braw, float* __restrict__ ssc) {
  const int img = blockIdx.x;
  const int tid = threadIdx.x;
  __shared__ unsigned       skey[SORTN];   // 32 KB
  __shared__ unsigned short sidx[SORTN];   // 16 KB

  const float* sIn = selScore + (size_t)img * KPRE;

  for (int m = 0; m < 5; ++m) {
    int s = tid + (m << 10);
    if (s < KPRE) {
      unsigned lds_off = (unsigned)(unsigned long long)(&skey[s]);
      const float* gp = sIn + s;
      asm volatile("global_load_async_to_lds_b32 %0, %1, off"
                   :: "v"(lds_off), "v"(gp) : "memory");
    }
  }
  for (int s = tid; s < SORTN; s += 1024) {
    if (s >= KPRE) skey[s] = 0u;
    sidx[s] = (unsigned short)s;
  }
  asm volatile("s_wait_asynccnt 0" ::: "memory");
  __syncthreads();
  for (int s = tid; s < KPRE; s += 1024)
    skey[s] = f2key(__uint_as_float(skey[s]));

  for (unsigned k2 = 2; k2 <= SORTN; k2 <<= 1) {
    for (unsigned j = k2 >> 1; j > 0; j >>= 1) {
      __syncthreads();
      for (int m = 0; m < 8; ++m) {
        unsigned i = (unsigned)tid + ((unsigned)m << 10);
        unsigned p = i ^ j;
        if (p > i) {
          unsigned ki = skey[i], kp = skey[p];
          bool desc = ((i & k2) == 0u);
          if (desc ? (ki < kp) : (ki > kp)) {
            skey[i] = kp; skey[p] = ki;
            unsigned short t = sidx[i]; sidx[i] = sidx[p]; sidx[p] = t;
          }
        }
      }
    }
  }
  __syncthreads();

  const int*    iIn  = selIdx + (size_t)img * KPRE;
  const float4* anc4 = (const float4*)anchors;
  const float4* dl4  = (const float4*)deltas + (size_t)img * ATOT;
  for (int m = 0; m < 5; ++m) {
    int s = tid + (m << 10);
    if (s >= KPRE) continue;
    int slot = (int)sidx[s];
    int a = iIn[slot];
    float score = sIn[slot];
    float4 an = anc4[a];
    float4 dl = dl4[a];
    float w  = an.z - an.x, h  = an.w - an.y;
    float cx = an.x + 0.5f * w, cy = an.y + 0.5f * h;
    float dw = fminf(dl.z, BBOX_CLIPV), dh = fminf(dl.w, BBOX_CLIPV);
    float pcx = dl.x * w + cx, pcy = dl.y * h + cy;
    float pw = expf(dw) * w,  ph = expf(dh) * h;
    float x1 = pcx - 0.5f * pw, y1 = pcy - 0.5f * ph;
    float x2 = pcx + 0.5f * pw, y2 = pcy + 0.5f * ph;
    x1 = fminf(fmaxf(x1, 0.0f), IMGW);
    y1 = fminf(fmaxf(y1, 0.0f), IMGH);
    x2 = fminf(fmaxf(x2, 0.0f), IMGW);
    y2 = fminf(fmaxf(y2, 0.0f), IMGH);
    bool valid = ((x2 - x1) >= 1e-3f) && ((y2 - y1) >= 1e-3f);
    int lvl = slot / 1000; if (lvl > 4) lvl = 4;
    float o = (float)lvl * 1217.0f;
    size_t d = (size_t)img * KPRE + s;
    braw[d] = make_float4(x1, y1, x2, y2);
    bnms[d] = make_float4(x1 + o, y1 + o, x2 + o, y2 + o);
    ssc[d]  = valid ? score : -__builtin_inff();
  }
}

// ============================================================================
// Kernel 3a: suppression bitmask generation (fully parallel).
// Block = (col-chunk, row-tile, image); 256 threads = 256 rows.
// Column boxes for the chunk are staged into LDS with async b128 copies.
// Mask is 22.7 MB total -> stays resident in the 192 MB L2.
// ============================================================================
__global__ __launch_bounds__(RTILE) void rpn_nms_mask(
    const float4* __restrict__ bnms, unsigned long long* __restrict__ mask) {
  const int chunk = blockIdx.x, rowT = blockIdx.y, img = blockIdx.z;
  const int tid = threadIdx.x;
  const float4* bn = bnms + (size_t)img * KPRE;

  __shared__ float4 cb[WPC * 64];            // 960 column boxes, 15 KB
  for (int c = tid; c < WPC * 64; c += RTILE) {
    int j = chunk * (WPC * 64) + c;
    if (j < KPRE) {
      unsigned lds_off = (unsigned)(unsigned long long)(&cb[c]);
      asm volatile("global_load_async_to_lds_b128 %0, %1, off"
                   :: "v"(lds_off), "v"(bn + j) : "memory");
    } else {
      cb[c] = make_float4(0.0f, 0.0f, 0.0f, 0.0f);
    }
  }
  asm volatile("s_wait_asynccnt 0" ::: "memory");
  __syncthreads();

  const int i = rowT * RTILE + tid;          // row index
  if (i >= KPRE) return;
  float4 bi = bn[i];
  float ai = (bi.z - bi.x) * (bi.w - bi.y);

  unsigned long long* mrow = mask + ((size_t)img * KPRE + i) * CWORDS;
  for (int w = 0; w < WPC; ++w) {
    const int gw = chunk * WPC + w;          // global word, 0..74
    unsigned long long bits = 0ull;
    const int jbase = gw * 64;
    if (jbase + 63 > i) {                    // word can contain j > i
      #pragma unroll 4
      for (int b = 0; b < 64; ++b) {
        int j = jbase + b;
        if (j > i && j < KPRE) {
          float4 bj = cb[w * 64 + b];
          float lx = fmaxf(bi.x, bj.x), ly = fmaxf(bi.y, bj.y);
          float rx = fminf(bi.z, bj.z), ry = fminf(bi.w, bj.w);
          float iw = fmaxf(rx - lx, 0.0f), ih = fmaxf(ry - ly, 0.0f);
          float inter = iw * ih;
          float aj = (bj.z - bj.x) * (bj.w - bj.y);
          float iou = inter / (ai + aj - inter + 1e-9f);
          if (iou > NMS_TH) bits |= (1ull << b);
        }
      }
    }
    mrow[gw] = bits;
  }
}

// ============================================================================
// Kernel 3b: sequential greedy reduce -- ONE wave32 per image, zero barriers.
// remv[75] lives in LDS; each lane owns words lane, lane+32, lane+64.
// Single-wave LDS ops are in order, so iteration i+1 sees iteration i's ORs.
// Mask rows are prefetched ahead so the OR-update hits cache.
// ============================================================================
__global__ __launch_bounds__(32) void rpn_nms_reduce(
    const unsigned long long* __restrict__ mask, const float* __restrict__ ssc,
    unsigned char* __restrict__ keepOut) {
  const int img = blockIdx.x;
  const int lane = threadIdx.x;
  __shared__ unsigned long long remv[CWORDS];       // 600 B
  __shared__ float lsc[KPRE];                        // 19 KB
  __shared__ unsigned char kflag[KPRE];              // 4.7 KB

  const unsigned long long* M = mask + (size_t)img * KPRE * CWORDS;
  const float* sc = ssc + (size_t)img * KPRE;

  for (int w = lane; w < CWORDS; w += 32) remv[w] = 0ull;
  for (int s = lane; s < KPRE; s += 32) lsc[s] = sc[s];
  __builtin_amdgcn_wave_barrier();

  for (int i = 0; i < KPRE; ++i) {
    __builtin_prefetch(M + (size_t)(i + 8) * CWORDS + lane * 3, 0, 1);
    const int w = i >> 6, b = i & 63;
    unsigned long long rw = remv[w];                 // uniform LDS read
    bool kept = (((rw >> b) & 1ull) == 0ull) && (lsc[i] > -3.0e38f);
    kflag[i] = kept ? (unsigned char)1 : (unsigned char)0;
    if (kept) {
      const unsigned long long* row = M + (size_t)i * CWORDS;
      #pragma unroll
      for (int c = 0; c < 3; ++c) {
        int wi = lane + (c << 5);
        if (wi < CWORDS) remv[wi] |= row[wi];
      }
    }
    __builtin_amdgcn_wave_barrier();
  }

  unsigned char* ko = keepOut + (size_t)img * KPRE;
  for (int s = lane; s < KPRE; s += 32) ko[s] = kflag[s];
}

// ============================================================================
// Kernel 3c: stable compaction + output (score order preserved), zero-fill.
// ============================================================================
__global__ __launch_bounds__(1024) void rpn_output(
    const float4* __restrict__ braw, const float* __restrict__ ssc,
    const unsigned char* __restrict__ keepIn, float* __restrict__ out) {
  const int img = blockIdx.x;
  const int tid = threadIdx.x;
  __shared__ unsigned char keep[KPRE];
  __shared__ unsigned cnt[1024];
  __shared__ unsigned excl[1024];
  __shared__ unsigned sh_total;

  const unsigned char* ki = keepIn + (size_t)img * KPRE;
  for (int s = tid; s < KPRE; s += 1024) keep[s] = ki[s];
  __syncthreads();

  unsigned c = 0;
  const int base_p = tid * 5;
  for (int l = 0; l < 5; ++l) {
    int p = base_p + l;
    if (p < KPRE && keep[p]) ++c;
  }
  cnt[tid] = c;
  __syncthreads();
  if (tid == 0) {
    unsigned run = 0;
    for (int t = 0; t < 1024; ++t) { excl[t] = run; run += cnt[t]; }
    sh_total = run;
  }
  __syncthreads();

  const float4* br = braw + (size_t)img * KPRE;
  const float*  sc = ssc  + (size_t)img * KPRE;
  float* o = out + (size_t)img * POSTN * 5;
  unsigned rank = excl[tid];
  for (int l = 0; l < 5; ++l) {
    int p = base_p + l;
    if (p < KPRE && keep[p]) {
      if (rank < POSTN) {
        float4 b = br[p];
        float sv = sc[p];
        float* w = o + (size_t)rank * 5;
        w[0] = b.x; w[1] = b.y; w[2] = b.z; w[3] = b.w; w[4] = sv;
      }
      ++rank;
    }
  }
  unsigned total = sh_total; if (total > POSTN) total = POSTN;
  for (unsigned r = tid; r < POSTN; r += 1024) {
    if (r >= total) {
      float* w = o + (size_t)r * 5;
      w[0] = 0.0f; w[1] = 0.0f; w[2] = 0.0f; w[3] = 0.0f; w[4] = 0.0f;
    }
  }
}

// ============================================================================
// Host launcher. Workspace layout (bytes), total ~24.5 MB (mask fits in L2):
//   [0,        151712)   selScore : 8*4741 f32
//   [151712,   303424)   selIdx   : 8*4741 i32
//   [303424,   910272)   bnms     : 8*4741 float4 (NMS coords, level-offset)
//   [910272,  1517120)   braw     : 8*4741 float4 (clipped coords)
//   [1517120, 1668832)   ssc      : 8*4741 f32 (sorted scores, -inf if small)
//   [1668832, 1706760)   keep     : 8*4741 u8
//   [1706816, 24463616)  mask     : 8*4741*75 u64 suppression bitmask
// ============================================================================
extern "C" void kernel_launch(void* const* d_in, const int* in_sizes, int n_in,
                              void* d_out, int out_size, void* d_ws, size_t ws_size,
                              hipStream_t stream) {
  (void)in_sizes; (void)n_in; (void)out_size; (void)ws_size;
  const float* obj     = (const float*)d_in[0];   // [8, 242991]
  const float* deltas  = (const float*)d_in[1];   // [8, 242991, 4]
  const float* anchors = (const float*)d_in[2];   // [242991, 4]
  float* out = (float*)d_out;                     // [8, 1000, 5]

  char* ws = (char*)d_ws;
  float*  selScore = (float*)(ws);
  int*    selIdx   = (int*)  (ws + 151712);
  float4* bnms     = (float4*)(ws + 303424);
  float4* braw     = (float4*)(ws + 910272);
  float*  ssc      = (float*)(ws + 1517120);
  unsigned char* keep = (unsigned char*)(ws + 1668832);
  unsigned long long* mask = (unsigned long long*)(ws + 1706816);

  rpn_topk<<<NIMG * 5, 1024, 0, stream>>>(obj, selScore, selIdx);
  rpn_sort_decode<<<NIMG, 1024, 0, stream>>>(selScore, selIdx, deltas, anchors,
                                             bnms, braw, ssc);
  dim3 mg(NCHUNK, (KPRE + RTILE - 1) / RTILE, NIMG);   // 5 x 19 x 8 = 760 blocks
  rpn_nms_mask<<<mg, RTILE, 0, stream>>>(bnms, mask);
  rpn_nms_reduce<<<NIMG, 32, 0, stream>>>(mask, ssc, keep);
  rpn_output<<<NIMG, 1024, 0, stream>>>(braw, ssc, keep, out);
}